// GraphTransformer_52956946760184
// MI455X (gfx1250) — compile-verified
//
#include <hip/hip_runtime.h>
#include <hip/hip_bf16.h>
#include <math.h>

#define N_NODES 20000
#define N_EDGES 640000
#define DIM     128
#define NGRAPH  64
#define NCLS    10

typedef float v2f __attribute__((ext_vector_type(2)));
typedef float v8f __attribute__((ext_vector_type(8)));

// ---------------------------------------------------------------------------
// WMMA fp32 GEMM: C[M,128] = A[M,128] @ W[128,128] + bias (row-major W).
// One wave computes a 16x64 strip = 4 adjacent 16x16 tiles; the A fragment
// (one global_load_b64 per K-step) feeds 4x V_WMMA_F32_16X16X4_F32.
// Column stride is the compile-time constant DIM so all K-loop loads use
// immediate IOFFSETs off fixed base pointers (no per-iteration address math).
// A 16x4 layout: lanes 0-15 -> M=lane, K={k,k+1}; lanes 16-31 -> K={k+2,k+3}.
// B 4x16 layout: lanes 0-15 -> N=lane, K rows {k,k+1}; lanes 16-31 -> {k+2,k+3}.
// C 16x16 layout: VGPR j -> row j (lanes 0-15) / row j+8 (lanes 16-31).
// ---------------------------------------------------------------------------
__global__ void gemm_wmma_f32(const float* __restrict__ A,
                              const float* __restrict__ W,
                              const float* __restrict__ bias,
                              float* __restrict__ C, int M) {
  const int wave = (int)((blockIdx.x * blockDim.x + threadIdx.x) >> 5);
  const int lane = (int)(threadIdx.x & 31);
  const int mTile = wave >> 1;              // two 64-col strips per 16-row tile
  const int strip = wave & 1;               // 0 -> cols 0..63, 1 -> cols 64..127
  if (mTile * 16 >= M) return;              // whole-wave exit only (EXEC stays all-1s)

  const int l15  = lane & 15;
  const int kOff = (lane >> 4) << 1;        // 0 (lanes 0-15) or 2 (lanes 16-31)
  const float* __restrict__ arow  = A + (size_t)(mTile * 16 + l15) * DIM + kOff;
  const float* __restrict__ wbase = W + (size_t)kOff * DIM + strip * 64 + l15;

  v8f acc0 = {}, acc1 = {}, acc2 = {}, acc3 = {};
#pragma unroll
  for (int k = 0; k < DIM; k += 4) {
    const v2f a = *(const v2f*)(arow + k);
    const float* wk = wbase + (size_t)k * DIM;
    v2f b0, b1, b2, b3;
    b0.x = wk[0];   b0.y = wk[DIM];
    b1.x = wk[16];  b1.y = wk[DIM + 16];
    b2.x = wk[32];  b2.y = wk[DIM + 32];
    b3.x = wk[48];  b3.y = wk[DIM + 48];
    acc0 = __builtin_amdgcn_wmma_f32_16x16x4_f32(false, a, false, b0, (short)0, acc0, false, false);
    acc1 = __builtin_amdgcn_wmma_f32_16x16x4_f32(false, a, false, b1, (short)0, acc1, false, false);
    acc2 = __builtin_amdgcn_wmma_f32_16x16x4_f32(false, a, false, b2, (short)0, acc2, false, false);
    acc3 = __builtin_amdgcn_wmma_f32_16x16x4_f32(false, a, false, b3, (short)0, acc3, false, false);
  }

  const int colBase = strip * 64 + l15;
  const float bc0 = bias[colBase];
  const float bc1 = bias[colBase + 16];
  const float bc2 = bias[colBase + 32];
  const float bc3 = bias[colBase + 48];
  float* __restrict__ crow =
      C + (size_t)(mTile * 16 + ((lane >> 4) << 3)) * DIM + colBase;
#pragma unroll
  for (int j = 0; j < 8; ++j) {
    crow[(size_t)j * DIM +  0] = acc0[j] + bc0;
    crow[(size_t)j * DIM + 16] = acc1[j] + bc1;
    crow[(size_t)j * DIM + 32] = acc2[j] + bc2;
    crow[(size_t)j * DIM + 48] = acc3[j] + bc3;
  }
}

// --------------------------- edge attention ---------------------------------
__device__ __forceinline__ unsigned f2mono(float f) {
  unsigned u = __float_as_uint(f);
  return (u & 0x80000000u) ? ~u : (u | 0x80000000u);
}
__device__ __forceinline__ float mono2f(unsigned m) {
  unsigned u = (m & 0x80000000u) ? (m & 0x7fffffffu) : ~m;
  return __uint_as_float(u);
}

// one wave per edge: alpha[e] = dot(q[dst], k[src]) / sqrt(D)
__global__ void edge_scores(const float* __restrict__ q, const float* __restrict__ k,
                            const int* __restrict__ src, const int* __restrict__ dst,
                            float* __restrict__ alpha, int E) {
  const int e    = (int)((blockIdx.x * blockDim.x + threadIdx.x) >> 5);
  const int lane = (int)(threadIdx.x & 31);
  if (e >= E) return;
  const int s = src[e], d = dst[e];
  const float4 qa = ((const float4*)(q + (size_t)d * DIM))[lane];
  const float4 ka = ((const float4*)(k + (size_t)s * DIM))[lane];
  float p = qa.x * ka.x + qa.y * ka.y + qa.z * ka.z + qa.w * ka.w;
#pragma unroll
  for (int off = 16; off; off >>= 1) p += __shfl_xor(p, off, 32);
  if (lane == 0) alpha[e] = p * 0.08838834764831845f;   // 1/sqrt(128)
}

__global__ void seg_max_kernel(const float* __restrict__ alpha,
                               const int* __restrict__ dst,
                               unsigned* __restrict__ mkey, int E) {
  const int e = blockIdx.x * blockDim.x + threadIdx.x;
  if (e >= E) return;
  atomicMax(&mkey[dst[e]], f2mono(alpha[e]));
}

__global__ void exp_sum_kernel(const float* __restrict__ alpha,
                               const int* __restrict__ dst,
                               const unsigned* __restrict__ mkey,
                               float* __restrict__ expv, float* __restrict__ ssum, int E) {
  const int e = blockIdx.x * blockDim.x + threadIdx.x;
  if (e >= E) return;
  const int d = dst[e];
  const float ex = expf(alpha[e] - mono2f(mkey[d]));
  expv[e] = ex;
  atomicAdd(&ssum[d], ex);
}

// one wave per edge: aggr[dst] += (expv/ssum) * v[src]
__global__ void scatter_kernel(const float* __restrict__ expv, const float* __restrict__ ssum,
                               const float* __restrict__ v,
                               const int* __restrict__ src, const int* __restrict__ dst,
                               float* __restrict__ aggr, int E) {
  const int e    = (int)((blockIdx.x * blockDim.x + threadIdx.x) >> 5);
  const int lane = (int)(threadIdx.x & 31);
  if (e >= E) return;
  const int s = src[e], d = dst[e];
  const float w = expv[e] / (ssum[d] + 1e-16f);
  const float4 vv = ((const float4*)(v + (size_t)s * DIM))[lane];
  float* o = aggr + (size_t)d * DIM + lane * 4;
  atomicAdd(o + 0, w * vv.x);
  atomicAdd(o + 1, w * vv.y);
  atomicAdd(o + 2, w * vv.z);
  atomicAdd(o + 3, w * vv.w);
}

__global__ void combine_kernel(const float* __restrict__ aggr, const float* __restrict__ skip,
                               float* __restrict__ out, int n, int relu) {
  const int i = blockIdx.x * blockDim.x + threadIdx.x;
  if (i >= n) return;
  const float t = aggr[i] + skip[i];
  out[i] = relu ? fmaxf(t, 0.0f) : t;
}

// ------------------------------- head ---------------------------------------
__global__ void pool_kernel(const float* __restrict__ h, const int* __restrict__ batch,
                            float* __restrict__ g) {
  const int i = blockIdx.x * blockDim.x + threadIdx.x;
  if (i >= N_NODES * DIM) return;
  const int n = i >> 7, c = i & (DIM - 1);
  atomicAdd(&g[(size_t)batch[n] * DIM + c], h[i]);
}

__global__ void bn_relu_kernel(const float* __restrict__ g1,
                               const float* __restrict__ gamma, const float* __restrict__ beta,
                               float* __restrict__ g2) {
  const int c = threadIdx.x;          // 128 threads, one per channel
  float s = 0.f, s2 = 0.f;
  for (int r = 0; r < NGRAPH; ++r) { const float t = g1[r * DIM + c]; s += t; s2 += t * t; }
  const float mu  = s / (float)NGRAPH;
  const float var = s2 / (float)NGRAPH - mu * mu;   // biased var, matches reference
  const float inv = rsqrtf(var + 1e-5f);
  const float ga = gamma[c], be = beta[c];
  for (int r = 0; r < NGRAPH; ++r) {
    const float t = (g1[r * DIM + c] - mu) * inv * ga + be;
    g2[r * DIM + c] = fmaxf(t, 0.0f);
  }
}

// one wave per graph: logits + log_softmax over 10 classes
__global__ void head_kernel(const float* __restrict__ g2, const float* __restrict__ W2,
                            const float* __restrict__ b2, float* __restrict__ out) {
  const int row  = (int)((blockIdx.x * blockDim.x + threadIdx.x) >> 5);
  const int lane = (int)(threadIdx.x & 31);
  if (row >= NGRAPH) return;
  float logit = -INFINITY;
  if (lane < NCLS) {
    float acc = b2[lane];
    for (int k = 0; k < DIM; ++k) acc += g2[(size_t)row * DIM + k] * W2[k * NCLS + lane];
    logit = acc;
  }
  float m = logit;
#pragma unroll
  for (int off = 16; off; off >>= 1) m = fmaxf(m, __shfl_xor(m, off, 32));
  float ex = (lane < NCLS) ? expf(logit - m) : 0.0f;
  float ssum = ex;
#pragma unroll
  for (int off = 16; off; off >>= 1) ssum += __shfl_xor(ssum, off, 32);
  if (lane < NCLS) out[row * NCLS + lane] = (logit - m) - logf(ssum);
}

// ---------------------------------------------------------------------------
extern "C" void kernel_launch(void* const* d_in, const int* in_sizes, int n_in,
                              void* d_out, int out_size, void* d_ws, size_t ws_size,
                              hipStream_t stream) {
  (void)in_sizes; (void)n_in; (void)out_size; (void)ws_size;

  const float* x    = (const float*)d_in[0];
  const int*  eidx  = (const int*)d_in[1];     // edge_index flattened [2,E] (int32 per harness)
  const int*  src   = eidx;
  const int*  dst   = eidx + N_EDGES;
  const int*  batch = (const int*)d_in[2];
  const float* P[38];
  for (int i = 0; i < 38; ++i) P[i] = (const float*)d_in[3 + i];
  // per layer l: P[8l+0..7] = Wq,bq,Wk,bk,Wv,bv,Wskip,bskip
  // P[32]=W_lin1 P[33]=b_lin1 P[34]=bn_gamma P[35]=bn_beta P[36]=W_lin2 P[37]=b_lin2

  char* ws = (char*)d_ws;
  const size_t NB = (size_t)N_NODES * DIM * sizeof(float);
  const size_t EB = (size_t)N_EDGES * sizeof(float);
  float*    hA    = (float*)(ws + 0 * NB);
  float*    hB    = (float*)(ws + 1 * NB);
  float*    qb    = (float*)(ws + 2 * NB);
  float*    kb    = (float*)(ws + 3 * NB);
  float*    vb    = (float*)(ws + 4 * NB);
  float*    skb   = (float*)(ws + 5 * NB);
  float*    agg   = (float*)(ws + 6 * NB);
  float*    alpha = (float*)(ws + 7 * NB);
  float*    expv  = (float*)(ws + 7 * NB + EB);
  unsigned* mkey  = (unsigned*)(ws + 7 * NB + 2 * EB);
  float*    ssum  = (float*)(ws + 7 * NB + 2 * EB + (size_t)N_NODES * 4);
  float*    gpool = (float*)(ws + 7 * NB + 2 * EB + 2 * (size_t)N_NODES * 4);
  float*    g1    = gpool + NGRAPH * DIM;
  float*    g2    = g1 + NGRAPH * DIM;

  // GEMM: 2 waves (16x64 strips) per 16-row tile, 8 waves per 256-thread block
  const int gemmWavesN  = (N_NODES / 16) * 2;                    // 2500 waves
  const int gemmBlocksN = (gemmWavesN + 7) / 8;                  // 313 blocks
  const int gemmWavesG  = (NGRAPH / 16) * 2;                     // 8 waves
  const int gemmBlocksG = (gemmWavesG + 7) / 8;                  // 1 block
  const int edgeBlocks  = N_EDGES / 8;                           // wave-per-edge, 8 waves/blk
  const int eltBlocksE  = (N_EDGES + 255) / 256;
  const int eltBlocksND = (N_NODES * DIM + 255) / 256;

  const float* hcur = x;
  for (int l = 0; l < 4; ++l) {
    const float *Wq = P[8*l+0], *bq = P[8*l+1], *Wk = P[8*l+2], *bk = P[8*l+3],
                *Wv = P[8*l+4], *bv = P[8*l+5], *Wsk = P[8*l+6], *bsk = P[8*l+7];

    hipMemsetAsync(agg,  0, NB, stream);
    hipMemsetAsync(mkey, 0, (size_t)N_NODES * 4, stream);
    hipMemsetAsync(ssum, 0, (size_t)N_NODES * 4, stream);

    gemm_wmma_f32<<<gemmBlocksN, 256, 0, stream>>>(hcur, Wq,  bq,  qb,  N_NODES);
    gemm_wmma_f32<<<gemmBlocksN, 256, 0, stream>>>(hcur, Wk,  bk,  kb,  N_NODES);
    gemm_wmma_f32<<<gemmBlocksN, 256, 0, stream>>>(hcur, Wv,  bv,  vb,  N_NODES);
    gemm_wmma_f32<<<gemmBlocksN, 256, 0, stream>>>(hcur, Wsk, bsk, skb, N_NODES);

    edge_scores   <<<edgeBlocks, 256, 0, stream>>>(qb, kb, src, dst, alpha, N_EDGES);
    seg_max_kernel<<<eltBlocksE, 256, 0, stream>>>(alpha, dst, mkey, N_EDGES);
    exp_sum_kernel<<<eltBlocksE, 256, 0, stream>>>(alpha, dst, mkey, expv, ssum, N_EDGES);
    scatter_kernel<<<edgeBlocks, 256, 0, stream>>>(expv, ssum, vb, src, dst, agg, N_EDGES);

    float* hnext = (l & 1) ? hB : hA;
    combine_kernel<<<eltBlocksND, 256, 0, stream>>>(agg, skb, hnext, N_NODES * DIM, (l < 3) ? 1 : 0);
    hcur = hnext;
  }

  hipMemsetAsync(gpool, 0, (size_t)NGRAPH * DIM * 4, stream);
  pool_kernel<<<eltBlocksND, 256, 0, stream>>>(hcur, batch, gpool);

  gemm_wmma_f32<<<gemmBlocksG, 256, 0, stream>>>(gpool, P[32], P[33], g1, NGRAPH);
  bn_relu_kernel<<<1, 128, 0, stream>>>(g1, P[34], P[35], g2);
  head_kernel<<<NGRAPH / 8, 256, 0, stream>>>(g2, P[36], P[37], (float*)d_out);
}